// HierarchicalContrastiveLoss_14791867367489
// MI455X (gfx1250) — compile-verified
//
#include <hip/hip_runtime.h>
#include <math.h>

typedef __attribute__((ext_vector_type(2))) float v2f;
typedef __attribute__((ext_vector_type(4))) float v4f;
typedef __attribute__((ext_vector_type(8))) float v8f;

#define BS    256
#define DIM   512
#define NTHR  256
#define NWAVE (NTHR / 32)
#define INV_TEMP (1.0f / 0.07f)

// ---------------- reduction helpers (wave32-aware) ----------------

__device__ __forceinline__ void waveReduceSum2(float& a, float& b) {
#pragma unroll
  for (int off = 16; off > 0; off >>= 1) {
    a += __shfl_xor(a, off, 32);
    b += __shfl_xor(b, off, 32);
  }
}

__device__ __forceinline__ v2f blockReduceSum2(float a, float b, float* scratch) {
  waveReduceSum2(a, b);
  const int wid = threadIdx.x >> 5;
  if ((threadIdx.x & 31) == 0) { scratch[2 * wid] = a; scratch[2 * wid + 1] = b; }
  __syncthreads();
  float ra = 0.f, rb = 0.f;
#pragma unroll
  for (int i = 0; i < NWAVE; ++i) { ra += scratch[2 * i]; rb += scratch[2 * i + 1]; }
  __syncthreads();
  v2f r; r.x = ra; r.y = rb;
  return r;
}

__device__ __forceinline__ float blockReduceMax(float a, float* scratch) {
#pragma unroll
  for (int off = 16; off > 0; off >>= 1)
    a = fmaxf(a, __shfl_xor(a, off, 32));
  const int wid = threadIdx.x >> 5;
  if ((threadIdx.x & 31) == 0) scratch[wid] = a;
  __syncthreads();
  float r = -INFINITY;
#pragma unroll
  for (int i = 0; i < NWAVE; ++i) r = fmaxf(r, scratch[i]);
  __syncthreads();
  return r;
}

// ---------------- fused per-batch normalize-mean + attention-pool ----------------
// One block per batch element, 8 waves. WAVE-PER-FRAME: wave w owns frames
// w, w+8, ...; lane owns 16 contiguous dims (4x b128 loads, wave covers a
// contiguous 2KB row). The per-frame (sumsq, score) reduction is a pure
// shuffle wave-reduce -> the HBM streaming loop has ZERO block barriers.
// Per-wave partial mean/pool vectors are combined through LDS once.
// Pass 2 re-reads rows from L2 (block slice <= 0.4 MB, global L2 = 192 MB).
__global__ __launch_bounds__(NTHR)
void pool_norm_kernel(const float* __restrict__ seq, const float* __restrict__ w,
                      const float* __restrict__ bias, int L,
                      float* __restrict__ mean_out, float* __restrict__ pool_out) {
  __shared__ float s_part[2 * NWAVE];
  __shared__ float s_scores[256];                    // L <= 196
  __shared__ __align__(16) float s_acc[NWAVE][DIM];  // 16KB, reused pass1/pass2

  const int b    = blockIdx.x;
  const int t    = threadIdx.x;
  const int wave = t >> 5;
  const int lane = t & 31;
  const float* base = seq + (size_t)b * (size_t)L * DIM;
  const float bb = bias[0];

  // per-lane chunk of the score weight vector
  v4f wv4[4];
#pragma unroll
  for (int i = 0; i < 4; ++i) wv4[i] = *(const v4f*)(w + lane * 16 + 4 * i);

  // ---- pass 1: stream frames; fused (sumsq, score) wave reduction ----
  v4f acc4[4];
#pragma unroll
  for (int i = 0; i < 4; ++i) acc4[i] = (v4f)0.f;

  for (int f = wave; f < L; f += NWAVE) {
    const float* row = base + (size_t)f * DIM + lane * 16;
    if (f + NWAVE < L) __builtin_prefetch(row + (size_t)NWAVE * DIM, 0, 1);
    v4f x[4];
#pragma unroll
    for (int i = 0; i < 4; ++i) x[i] = *(const v4f*)(row + 4 * i);
    float ss = 0.f, sc = 0.f;
#pragma unroll
    for (int i = 0; i < 4; ++i) {
      ss += x[i].x * x[i].x + x[i].y * x[i].y + x[i].z * x[i].z + x[i].w * x[i].w;
      sc += x[i].x * wv4[i].x + x[i].y * wv4[i].y + x[i].z * wv4[i].z + x[i].w * wv4[i].w;
    }
    waveReduceSum2(ss, sc);  // shuffle-only, no barrier
    const float inv = 1.0f / fmaxf(sqrtf(ss), 1e-12f);
#pragma unroll
    for (int i = 0; i < 4; ++i) acc4[i] += x[i] * inv;
    if (lane == 0) s_scores[f] = sc + bb;
  }

  // combine per-wave mean partials through LDS
#pragma unroll
  for (int i = 0; i < 4; ++i)
    *(v4f*)&s_acc[wave][lane * 16 + 4 * i] = acc4[i];
  __syncthreads();

  const float invL = 1.0f / (float)L;
  {
    float m0 = 0.f, m1 = 0.f;
#pragma unroll
    for (int wv = 0; wv < NWAVE; ++wv) {
      m0 += s_acc[wv][2 * t];
      m1 += s_acc[wv][2 * t + 1];
    }
    v2f mo; mo.x = m0 * invL; mo.y = m1 * invL;
    *(v2f*)(mean_out + (size_t)b * DIM + 2 * t) = mo;
  }

  // ---- softmax over the L scores ----
  float mloc = -INFINITY;
  for (int f = t; f < L; f += NTHR) mloc = fmaxf(mloc, s_scores[f]);
  const float mmax = blockReduceMax(mloc, s_part);  // barrier: s_acc reads done
  float sloc = 0.f;
  for (int f = t; f < L; f += NTHR) sloc += expf(s_scores[f] - mmax);
  const float invS = 1.0f / blockReduceSum2(sloc, 0.f, s_part).x;

  // ---- pass 2: wave-per-frame weighted sum (rows L2-hot) ----
  v4f p4[4];
#pragma unroll
  for (int i = 0; i < 4; ++i) p4[i] = (v4f)0.f;

  for (int f = wave; f < L; f += NWAVE) {
    const float a = expf(s_scores[f] - mmax) * invS;
    const float* row = base + (size_t)f * DIM + lane * 16;
    v4f x[4];
#pragma unroll
    for (int i = 0; i < 4; ++i) x[i] = *(const v4f*)(row + 4 * i);
#pragma unroll
    for (int i = 0; i < 4; ++i) p4[i] += x[i] * a;
  }

#pragma unroll
  for (int i = 0; i < 4; ++i)
    *(v4f*)&s_acc[wave][lane * 16 + 4 * i] = p4[i];
  __syncthreads();

  float p0 = 0.f, p1 = 0.f;
#pragma unroll
  for (int wv = 0; wv < NWAVE; ++wv) {
    p0 += s_acc[wv][2 * t];
    p1 += s_acc[wv][2 * t + 1];
  }
  // L2-normalize pooled vector
  v2f nrm = blockReduceSum2(p0 * p0 + p1 * p1, 0.f, s_part);
  const float inv = 1.0f / fmaxf(sqrtf(nrm.x), 1e-12f);
  v2f po; po.x = p0 * inv; po.y = p1 * inv;
  *(v2f*)(pool_out + (size_t)b * DIM + 2 * t) = po;
}

// ---------------- fp32 WMMA logits: C = (A @ B^T) / TEMP ----------------
// One wave per 16x16 tile; V_WMMA_F32_16X16X4_F32, K looped 512/4.
// A-frag: lane holds (M = lane&15, K = 2*(lane>>4)+v); B-frag identical with
// N = lane&15 — both read row-major [row][k] as float2 per step.
__global__ __launch_bounds__(NTHR)
void wmma_logits_kernel(const float* __restrict__ Am, const float* __restrict__ Bm,
                        const float* __restrict__ Ap, const float* __restrict__ Bp,
                        float* __restrict__ Cs, float* __restrict__ Cp) {
  const float* A; const float* B; float* C;
  if (blockIdx.y == 0) { A = Am; B = Bm; C = Cs; }
  else                 { A = Ap; B = Bp; C = Cp; }

  const int wave = threadIdx.x >> 5;
  const int lane = threadIdx.x & 31;
  const int tile = blockIdx.x * NWAVE + wave;  // 0..255
  const int tm = (tile >> 4) * 16;
  const int tn = (tile & 15) * 16;
  const int row = lane & 15;
  const int kh  = lane >> 4;  // 0/1: K half within the 4-wide step

  const float* aP = A + (size_t)(tm + row) * DIM + 2 * kh;
  const float* bP = B + (size_t)(tn + row) * DIM + 2 * kh;

  v8f c = {};
#pragma unroll 8
  for (int k = 0; k < DIM; k += 4) {
    v2f a = *(const v2f*)(aP + k);
    v2f b = *(const v2f*)(bP + k);
    c = __builtin_amdgcn_wmma_f32_16x16x4_f32(false, a, false, b, (short)0, c,
                                              false, false);
  }

  const int n = tn + row;
#pragma unroll
  for (int r = 0; r < 8; ++r) {
    const int m = tm + r + 8 * kh;  // D layout: VGPR r, lanes16-31 hold M=r+8
    C[(size_t)m * BS + n] = c[r] * INV_TEMP;
  }
}

// ---------------- bidirectional diag cross-entropy on both logit matrices ----------------
__global__ __launch_bounds__(NTHR)
void ce_kernel(const float* __restrict__ seqL, const float* __restrict__ poolL,
               float* __restrict__ out) {
  __shared__ float s_part[2 * NWAVE];
  const int i = threadIdx.x;
  float total = 0.f;
  for (int which = 0; which < 2; ++which) {
    const float* Lg = which ? poolL : seqL;
    float mr = -INFINITY, mc = -INFINITY;
    for (int j = 0; j < BS; ++j) {
      mr = fmaxf(mr, Lg[i * BS + j]);
      mc = fmaxf(mc, Lg[j * BS + i]);
    }
    float sr = 0.f, sc = 0.f;
    for (int j = 0; j < BS; ++j) {
      sr += expf(Lg[i * BS + j] - mr);
      sc += expf(Lg[j * BS + i] - mc);
    }
    const float diag = Lg[i * BS + i];
    const float contrib = (mr + logf(sr) - diag) + (mc + logf(sc) - diag);
    v2f red = blockReduceSum2(contrib, 0.f, s_part);
    total += 0.5f * red.x / (float)BS;  // per-matrix loss = 0.5*(lm+lt)
  }
  if (i == 0) out[0] = 0.5f * total;  // SEQ_W*seq + POOL_W*pool
}

// ---------------- launcher ----------------
extern "C" void kernel_launch(void* const* d_in, const int* in_sizes, int n_in,
                              void* d_out, int out_size, void* d_ws, size_t ws_size,
                              hipStream_t stream) {
  const float* motion = (const float*)d_in[0];  // [256,196,512]
  const float* text   = (const float*)d_in[1];  // [256, 77,512]
  const float* wm     = (const float*)d_in[2];  // [512]
  const float* bm     = (const float*)d_in[3];  // [1]
  const float* wt     = (const float*)d_in[4];  // [512]
  const float* bt     = (const float*)d_in[5];  // [1]

  float* ws     = (float*)d_ws;
  float* m_mean = ws;                    // 256*512
  float* t_mean = m_mean + BS * DIM;     // 256*512
  float* mp     = t_mean + BS * DIM;     // 256*512
  float* tp     = mp + BS * DIM;         // 256*512
  float* seqL   = tp + BS * DIM;         // 256*256
  float* poolL  = seqL + BS * BS;        // 256*256

  pool_norm_kernel<<<BS, NTHR, 0, stream>>>(motion, wm, bm, 196, m_mean, mp);
  pool_norm_kernel<<<BS, NTHR, 0, stream>>>(text,   wt, bt,  77, t_mean, tp);

  // 256 tiles of 16x16 over the 256x256 logits; 8 waves/block -> 32 blocks; y: seq/pooled
  wmma_logits_kernel<<<dim3(32, 2), NTHR, 0, stream>>>(m_mean, t_mean, mp, tp,
                                                       seqL, poolL);

  ce_kernel<<<1, NTHR, 0, stream>>>(seqL, poolL, (float*)d_out);
}